// RpnTarget_26800595927040
// MI455X (gfx1250) — compile-verified
//
#include <hip/hip_runtime.h>
#include <stdint.h>

// Problem constants (match reference)
#define BN     4
#define GT     64
#define NA     262144
#define TT     256
#define MAXPOS 230          // int(T*0.9)
#define K2     166          // MAXPOS - GT
#define CAP    4096         // candidate list capacity (anchors with IoU > 0.7)
#define BLK    256

// ---------- helpers ----------

// Order-preserving float->uint encoding (total order incl. negatives)
__device__ __forceinline__ unsigned ordf(float f) {
  unsigned u = __float_as_uint(f);
  return (u & 0x80000000u) ? ~u : (u | 0x80000000u);
}
__device__ __forceinline__ float unordf(unsigned e) {
  return (e & 0x80000000u) ? __uint_as_float(e ^ 0x80000000u)
                           : __uint_as_float(~e);
}

__device__ __forceinline__ float iou_one(float a0, float a1, float a2, float a3,
                                         float g0, float g1, float g2, float g3) {
  float y1 = fmaxf(g0, a0), x1 = fmaxf(g1, a1);
  float y2 = fminf(g2, a2), x2 = fminf(g3, a3);
  float inter = fmaxf(0.f, y2 - y1) * fmaxf(0.f, x2 - x1);
  float ag = (g2 - g0) * (g3 - g1);
  float aa = (a2 - a0) * (a3 - a1);
  float un = ag + aa - inter;
  return inter / (un > 0.f ? un : 1.f);
}

// CDNA5 async global->LDS DMA: one 16B chunk per active lane (ASYNCcnt path).
// VDST = per-lane LDS byte address, VADDR = per-lane byte offset, SADDR = base.
__device__ __forceinline__ void async_copy16(const void* gbase, unsigned goff,
                                             unsigned ldsaddr) {
  asm volatile("global_load_async_to_lds_b128 %0, %1, %2 offset:0"
               :: "v"(ldsaddr), "v"(goff),
                  "s"((unsigned long long)(uintptr_t)gbase)
               : "memory");
}
__device__ __forceinline__ void wait_async0() {
  asm volatile("s_wait_asynccnt 0" ::: "memory");
}

__device__ __forceinline__ unsigned long long shflxor_u64(unsigned long long v, int m) {
  unsigned lo = (unsigned)v, hi = (unsigned)(v >> 32);
  lo = __shfl_xor(lo, m, 32);
  hi = __shfl_xor(hi, m, 32);
  return ((unsigned long long)hi << 32) | lo;
}

// ---------- kernel 0: init workspace cells ----------
__global__ void k_init(unsigned long long* gt_pack, int* pos_cnt) {
  int i = blockIdx.x * blockDim.x + threadIdx.x;
  if (i < BN * GT) gt_pack[i] = 0ull;
  if (i < BN)      pos_cnt[i] = 0;
}

// ---------- kernel 1: IoU + reductions ----------
__global__ void __launch_bounds__(BLK)
k_iou(const float* __restrict__ gt_boxes,            // B*GT*5
      const unsigned char* __restrict__ tag,         // B*NA  (bool)
      const float* __restrict__ anchors,             // B*NA*4
      float* __restrict__ iou_max,                   // B*NA
      unsigned long long* __restrict__ gt_pack,      // B*GT  packed (iou, ~a)
      unsigned long long* __restrict__ cand,         // B*CAP packed (iou, ~a)
      int* __restrict__ pos_cnt)                     // B
{
  __shared__ float s_gtb[GT * 5];
  __shared__ unsigned long long s_key[GT];
  const int tid = threadIdx.x;
  const int b   = blockIdx.y;

  // Stage 64 GT boxes (320 floats = 80 x 16B) via async DMA to LDS.
  if (tid < (GT * 5 * 4) / 16) {
    unsigned lds = (unsigned)(uintptr_t)(&s_gtb[0]) + (unsigned)tid * 16u;
    async_copy16(gt_boxes + (size_t)b * GT * 5, (unsigned)tid * 16u, lds);
  }
  for (int g = tid; g < GT; g += BLK) s_key[g] = 0ull;
  wait_async0();
  __syncthreads();

  const int a = blockIdx.x * BLK + tid;
  const float4 av = ((const float4*)anchors)[(size_t)b * NA + a];
  const bool t = tag[(size_t)b * NA + a] != 0;

  float best = -1.0f;  // value of fully-masked row entries
  for (int g = 0; g < GT; ++g) {
    float g0 = s_gtb[g * 5 + 0], g1 = s_gtb[g * 5 + 1];
    float g2 = s_gtb[g * 5 + 2], g3 = s_gtb[g * 5 + 3];
    float gv = s_gtb[g * 5 + 4];
    float v = -1.0f;
    if (t && gv > 0.f) v = iou_one(av.x, av.y, av.z, av.w, g0, g1, g2, g3);
    if (v > best) best = v;  // strict > keeps first (smallest) g on ties

    // GT-side max over anchors: key = (ord(iou)<<32) | ~a  (tie -> smaller a)
    unsigned long long key =
        ((unsigned long long)ordf(v) << 32) | (unsigned)(~(unsigned)a);
    for (int off = 16; off > 0; off >>= 1) {
      unsigned long long o = shflxor_u64(key, off);
      if (o > key) key = o;
    }
    if ((tid & 31) == 0) atomicMax(&s_key[g], key);
  }

  iou_max[(size_t)b * NA + a] = best;
  if (best > 0.7f) {
    int p = atomicAdd(&pos_cnt[b], 1);
    if (p < CAP)
      cand[(size_t)b * CAP + p] =
          ((unsigned long long)ordf(best) << 32) | (unsigned)(~(unsigned)a);
  }
  __syncthreads();
  if (tid < GT) atomicMax(&gt_pack[(size_t)b * GT + tid], s_key[tid]);
}

// ---------- kernel 2: selection + output assembly (one block per batch) ----------
__global__ void __launch_bounds__(BLK)
k_assemble(const float* __restrict__ gt_boxes,       // B*GT*5
           const float* __restrict__ gt_cls,         // B*GT*2
           const float* __restrict__ anchors,        // B*NA*4
           const unsigned char* __restrict__ tag,    // B*NA
           const float* __restrict__ iou_max,        // B*NA
           const unsigned long long* __restrict__ gt_pack,
           const unsigned long long* __restrict__ cand,
           const int* __restrict__ pos_cnt,
           float* __restrict__ out)
{
  __shared__ float s_gtb[GT * 5];
  __shared__ float s_cls[GT * 2];
  __shared__ unsigned long long s_ck[CAP];
  __shared__ unsigned long long s_top[K2];
  __shared__ int s_negidx[TT];
  __shared__ int s_cnt[BLK + 1];
  __shared__ unsigned long long s_red[BLK / 32];
  __shared__ int s_redslot[BLK / 32];
  __shared__ int s_nvalid;

  const int tid = threadIdx.x;
  const int b   = blockIdx.x;

  // Async-stage GT boxes (80 x b128) and GT classes (32 x b128) to LDS.
  if (tid < 80) {
    unsigned lds = (unsigned)(uintptr_t)(&s_gtb[0]) + (unsigned)tid * 16u;
    async_copy16(gt_boxes + (size_t)b * GT * 5, (unsigned)tid * 16u, lds);
  } else if (tid < 112) {
    unsigned j = (unsigned)(tid - 80);
    unsigned lds = (unsigned)(uintptr_t)(&s_cls[0]) + j * 16u;
    async_copy16(gt_cls + (size_t)b * GT * 2, j * 16u, lds);
  }

  const int cnt = min(pos_cnt[b], CAP);
  for (int i = tid; i < CAP; i += BLK)
    s_ck[i] = (i < cnt) ? cand[(size_t)b * CAP + i] : 0ull;

  wait_async0();
  __syncthreads();

  // ---- negatives: first TT anchors (ascending index) with tag && max<0.3 ----
  const int CH = NA / BLK;  // 1024 contiguous anchors per thread
  const int base = tid * CH;
  int c = 0;
  for (int j = 0; j < CH; ++j) {
    int a = base + j;
    if (tag[(size_t)b * NA + a] && iou_max[(size_t)b * NA + a] < 0.3f) c++;
  }
  s_cnt[tid] = c;
  __syncthreads();
  if (tid == 0) {
    int acc = 0;
    for (int i = 0; i < BLK; ++i) { int v = s_cnt[i]; s_cnt[i] = acc; acc += v; }
    s_cnt[BLK] = acc;
    int nv = 0;
    for (int g = 0; g < GT; ++g) if (s_gtb[g * 5 + 4] > 0.f) nv++;
    s_nvalid = nv;
  }
  __syncthreads();
  {
    int rank = s_cnt[tid];
    if (rank < TT) {
      for (int j = 0; j < CH && rank < TT; ++j) {
        int a = base + j;
        if (tag[(size_t)b * NA + a] && iou_max[(size_t)b * NA + a] < 0.3f) {
          s_negidx[rank] = a;
          rank++;
        }
      }
    }
  }
  const int neg_total = s_cnt[BLK];
  const int nvalid = s_nvalid;

  // ---- top-K2 candidates by (iou desc, anchor idx asc): iterative extract ----
  for (int k = 0; k < K2; ++k) {
    unsigned long long bk = 0ull; int bs = -1;
    for (int i = tid; i < CAP; i += BLK) {
      unsigned long long v = s_ck[i];
      if (v > bk) { bk = v; bs = i; }
    }
    for (int off = 16; off > 0; off >>= 1) {
      unsigned long long ok = shflxor_u64(bk, off);
      int os = __shfl_xor(bs, off, 32);
      if (ok > bk) { bk = ok; bs = os; }
    }
    if ((tid & 31) == 0) { s_red[tid >> 5] = bk; s_redslot[tid >> 5] = bs; }
    __syncthreads();
    if (tid == 0) {
      unsigned long long m = 0ull; int ms = -1;
      for (int w = 0; w < BLK / 32; ++w)
        if (s_red[w] > m) { m = s_red[w]; ms = s_redslot[w]; }
      s_top[k] = m;
      if (ms >= 0) s_ck[ms] = 0ull;
    }
    __syncthreads();
  }

  const int cntk    = min(cnt, K2);
  const int pos_num = nvalid + cntk;           // valid GTs sorted first (stable)
  const int neg_num = min(TT - pos_num, neg_total);

  // ---- one output slot per thread ----
  const int i = tid;
  float d0 = 0.f, d1 = 0.f, d2 = 0.f, d3 = 0.f, dt = 0.f;
  float c0 = 0.f, c1 = 0.f, i0 = 0.f, i1 = 0.f;

  if (i < pos_num) {
    int g = 0, aa = 0;
    if (i < nvalid) {
      int cv = 0;
      for (int gg = 0; gg < GT; ++gg) {
        if (s_gtb[gg * 5 + 4] > 0.f) { if (cv == i) { g = gg; break; } cv++; }
      }
      unsigned long long pk = gt_pack[(size_t)b * GT + g];
      aa = (int)(~(unsigned)(pk & 0xFFFFFFFFull));
    } else {
      unsigned long long pk = s_top[i - nvalid];
      aa = (int)(~(unsigned)(pk & 0xFFFFFFFFull));
      // recompute per-anchor argmax over GTs (first max on ties)
      const float4 av4 = ((const float4*)anchors)[(size_t)b * NA + aa];
      float best = -2.f;
      for (int gg = 0; gg < GT; ++gg) {
        float gv = s_gtb[gg * 5 + 4];
        float v = (gv > 0.f)
            ? iou_one(av4.x, av4.y, av4.z, av4.w,
                      s_gtb[gg * 5 + 0], s_gtb[gg * 5 + 1],
                      s_gtb[gg * 5 + 2], s_gtb[gg * 5 + 3])
            : -1.f;
        if (v > best) { best = v; g = gg; }
      }
    }
    const float4 av = ((const float4*)anchors)[(size_t)b * NA + aa];
    float g0 = s_gtb[g * 5 + 0], g1 = s_gtb[g * 5 + 1];
    float g2 = s_gtb[g * 5 + 2], g3 = s_gtb[g * 5 + 3];
    float h = av.z - av.x, w = av.w - av.y;
    float gh = g2 - g0,   gw = g3 - g1;
    d0 = (g2 + g0 - (av.z + av.x)) * 0.5f / h * 10.f;   // / std 0.1
    d1 = (g3 + g1 - (av.w + av.y)) * 0.5f / w * 10.f;   // / std 0.1
    d2 = logf(gh / h) * 5.f;                            // / std 0.2
    d3 = logf(gw / w) * 5.f;                            // / std 0.2
    dt = 1.f;
    c0 = s_cls[g * 2 + 0]; c1 = 1.f;
    i0 = (float)aa;        i1 = 1.f;
  } else if (i < pos_num + neg_num) {
    dt = -1.f; c1 = -1.f; i1 = -1.f;
    i0 = (float)s_negidx[i - pos_num];
  }

  // Flat output layout: deltas(B,T,5) | class_ids(B,T,2) | indices(B,T,2) |
  //                     gt_num(B) | pos_num(B) | neg_num(B) | miss(B) | min_max_iou(B)
  float* outD = out;
  float* outC = out + (size_t)BN * TT * 5;
  float* outI = outC + (size_t)BN * TT * 2;
  float* outS = outI + (size_t)BN * TT * 2;
  size_t sl = (size_t)b * TT + i;
  outD[sl * 5 + 0] = d0; outD[sl * 5 + 1] = d1; outD[sl * 5 + 2] = d2;
  outD[sl * 5 + 3] = d3; outD[sl * 5 + 4] = dt;
  outC[sl * 2 + 0] = c0; outC[sl * 2 + 1] = c1;
  outI[sl * 2 + 0] = i0; outI[sl * 2 + 1] = i1;

  if (i == 0) {
    outS[0 * BN + b] = (float)nvalid;
    outS[1 * BN + b] = (float)pos_num;
    outS[2 * BN + b] = (float)neg_num;
    outS[3 * BN + b] = 0.f;  // every valid GT covered by its own argmax entry
    float mm = 1e9f;
    for (int g = 0; g < GT; ++g) {
      if (s_gtb[g * 5 + 4] > 0.f) {
        unsigned e = (unsigned)(gt_pack[(size_t)b * GT + g] >> 32);
        mm = fminf(mm, unordf(e));
      }
    }
    outS[4 * BN + b] = mm;
  }
}

// ---------- host-side launch ----------
extern "C" void kernel_launch(void* const* d_in, const int* in_sizes, int n_in,
                              void* d_out, int out_size, void* d_ws, size_t ws_size,
                              hipStream_t stream) {
  (void)in_sizes; (void)n_in; (void)out_size; (void)ws_size;
  const float* gt_boxes = (const float*)d_in[0];          // B*GT*5
  const float* gt_cls   = (const float*)d_in[1];          // B*GT*2
  const float* anchors  = (const float*)d_in[2];          // B*NA*4
  const unsigned char* tag = (const unsigned char*)d_in[3]; // B*NA bool
  float* out = (float*)d_out;

  // Workspace layout (~4.33 MB)
  float* iou_max = (float*)d_ws;                                        // B*NA
  unsigned long long* gt_pack =
      (unsigned long long*)(iou_max + (size_t)BN * NA);                 // B*GT
  unsigned long long* cand = gt_pack + (size_t)BN * GT;                 // B*CAP
  int* pos_cnt = (int*)(cand + (size_t)BN * CAP);                       // B

  k_init<<<1, BLK, 0, stream>>>(gt_pack, pos_cnt);

  dim3 g1(NA / BLK, BN);
  k_iou<<<g1, BLK, 0, stream>>>(gt_boxes, tag, anchors,
                                iou_max, gt_pack, cand, pos_cnt);

  k_assemble<<<BN, BLK, 0, stream>>>(gt_boxes, gt_cls, anchors, tag,
                                     iou_max, gt_pack, cand, pos_cnt, out);
}